// Wrapper_3547642986703
// MI455X (gfx1250) — compile-verified
//
#include <hip/hip_runtime.h>
#include <math.h>

typedef __attribute__((ext_vector_type(2))) float v2f;
typedef __attribute__((ext_vector_type(8))) float v8f;
typedef __attribute__((ext_vector_type(4))) unsigned int v4u;
typedef __attribute__((ext_vector_type(8))) int v8i;
typedef __attribute__((ext_vector_type(4))) int v4i;

#define CC    64        // channels
#define LTOT  4096      // H*W
#define KREAL 100       // clusters
#define KP    112       // clusters padded to 7*16
#define TL    128       // L tile per iteration
#define TLP   130       // s_x row stride in dwords (= TDM pad: 128 data + 2 pad)
#define CE    80        // extended channel dim (64 + ones row + 15 zero rows)
#define AS    132       // s_a row stride (TL + 4 pad)
#define AGS   81        // agg reuse stride
#define WSS   65        // s_w row stride (64 + 1 pad)
#define NT    (LTOT/TL) // 32 L-tiles

// Issue a TDM load of one [64 rows x 128 cols] fp32 tile (row stride 4096 elem)
// into LDS at lds_addr, with +2-dword padding after every 128-dword row.
__device__ __forceinline__ void tdm_issue_tile(unsigned lds_addr, const float* gptr) {
  unsigned long long ga = (unsigned long long)(uintptr_t)gptr;
  v4u g0 = { 1u,                                   // count=1, is_restore=0, gather=0
             lds_addr,                             // LDS byte address
             (unsigned)ga,                         // global_addr[31:0]
             (unsigned)((ga >> 32) & 0x01ffffffu) | 0x80000000u }; // addr[56:32] | type=2
  v8i g1 = { 0x03920000,   // data_size=4B(2<<16) | pad_en(1<<20) | pad_int=6(<<22) | pad_amt=1(<<25)
             0x10000000,   // tensor_dim0=4096 -> low16 in bits[63:48]
             0x00400000,   // tensor_dim0 hi16=0 | tensor_dim1=64 -> low16 in bits[95:80]
             0x00800000,   // tensor_dim1 hi16=0 | tile_dim0=128 in bits[127:112]
             0x00000040,   // tile_dim1=64, tile_dim2=0
             0x00001000,   // tensor_dim0_stride=4096 (low 32)
             0, 0 };       // stride hi, tensor_dim1_stride=0 (2-D)
  v4i z4 = { 0, 0, 0, 0 };
  v8i z8 = { 0, 0, 0, 0, 0, 0, 0, 0 };
  __builtin_amdgcn_tensor_load_to_lds(g0, g1, z4, z4, z8, 0);
}

__global__ __launch_bounds__(256) void netvlad_fused(
    const float* __restrict__ x, const float* __restrict__ conv_w,
    const float* __restrict__ conv_b, const float* __restrict__ cents,
    float* __restrict__ out)
{
  __shared__ float s_w[KP * WSS];      // conv weights, padded rows = 0
  __shared__ float s_b[KP];            // bias, padded rows = -1e30
  __shared__ float s_x[2][CE * TLP];   // double-buffered x tile [c][l]; row 64 = ones
  __shared__ float s_a[KP * AS];       // assignments; reused for agg at the end
  __shared__ float s_inv[TL];          // per-column 1/||x||
  __shared__ float s_gss;              // global sum of squares

  const int tid  = threadIdx.x;
  const int wv   = tid >> 5;
  const int lane = tid & 31;
  const int m16  = lane & 15;          // M-row within half / N-column
  const int half = lane >> 4;          // 0 or 1
  const int n    = blockIdx.x;

  // ---- one-time setup ----
  for (int idx = tid; idx < KP * CC; idx += 256) {
    int k = idx >> 6, c = idx & 63;
    s_w[k * WSS + c] = (k < KREAL) ? conv_w[k * CC + c] : 0.0f;
  }
  for (int idx = tid; idx < KP; idx += 256)
    s_b[idx] = (idx < KREAL) ? conv_b[idx] : -1.0e30f;
  // extended rows of both x buffers: row 64 = ones (asum trick), 65..79 = zeros
  for (int idx = tid; idx < 2 * 16 * TL; idx += 256) {
    int b = idx / (16 * TL);
    int r = idx - b * (16 * TL);
    int c = 64 + (r / TL);
    int l = r - (c - 64) * TL;
    s_x[b][c * TLP + l] = (c == 64) ? 1.0f : 0.0f;
  }
  if (tid == 0) s_gss = 0.0f;

  const unsigned xb_lds[2] = { (unsigned)(uintptr_t)&s_x[0][0],
                               (unsigned)(uintptr_t)&s_x[1][0] };
  const size_t xbase_n = (size_t)n * CC * LTOT;

  // prologue: DMA tile 0 into buffer 0 (wave 7 owns the TDM)
  if (wv == 7) tdm_issue_tile(xb_lds[0], x + xbase_n);

  v8f accA[5];
  #pragma unroll
  for (int ct = 0; ct < 5; ++ct) accA[ct] = (v8f){};

  for (int tile = 0; tile < NT; ++tile) {
    const int cur = tile & 1;
    const float* __restrict__ xc = &s_x[cur][0];

    if (wv == 7) {
      const int nxt = (tile + 1) & (NT - 1);          // tile 31 harmlessly re-fetches 0
      tdm_issue_tile(xb_lds[(tile + 1) & 1], x + xbase_n + (size_t)nxt * TL);
      __builtin_amdgcn_s_wait_tensorcnt(1);           // tile `tile` has landed
    }
    __syncthreads();   // (A) tile data + previous-iteration s_a readers done

    // ---- per-column inverse norms ----
    if (tid < TL) {
      float ss = 0.0f;
      #pragma unroll 8
      for (int c = 0; c < CC; ++c) { float v = xc[c * TLP + tid]; ss += v * v; }
      s_inv[tid] = 1.0f / fmaxf(sqrtf(ss), 1e-12f);
    }
    __syncthreads();   // (B)

    // ---- logits GEMM (W[112x64] * x[64x16]) + softmax; wave wv owns l-subtile wv ----
    {
      const int l0   = wv * 16;
      const int lcol = l0 + m16;
      const float myinv = s_inv[lcol];
      v8f lg[7];
      #pragma unroll
      for (int kt = 0; kt < 7; ++kt) {
        v8f acc = (v8f){};
        #pragma unroll
        for (int ks = 0; ks < 16; ++ks) {           // inner dim C=64, 4 per WMMA
          const int cb = ks * 4 + half * 2;
          v2f af, bf;
          af.x = s_w[(kt * 16 + m16) * WSS + cb];
          af.y = s_w[(kt * 16 + m16) * WSS + cb + 1];
          bf.x = xc[cb * TLP + lcol];
          bf.y = xc[(cb + 1) * TLP + lcol];
          acc = __builtin_amdgcn_wmma_f32_16x16x4_f32(
              false, af, false, bf, (short)0, acc, false, false);
        }
        #pragma unroll
        for (int j = 0; j < 8; ++j)
          acc[j] = acc[j] * myinv + s_b[kt * 16 + j + half * 8];
        lg[kt] = acc;
      }
      // softmax over K (rows split across lane-halves: combine via shfl_xor 16)
      float mx = -3.0e38f;
      #pragma unroll
      for (int kt = 0; kt < 7; ++kt)
        #pragma unroll
        for (int j = 0; j < 8; ++j) mx = fmaxf(mx, lg[kt][j]);
      mx = fmaxf(mx, __shfl_xor(mx, 16, 32));
      float sum = 0.0f;
      #pragma unroll
      for (int kt = 0; kt < 7; ++kt)
        #pragma unroll
        for (int j = 0; j < 8; ++j) {
          float e = __expf(lg[kt][j] - mx);
          lg[kt][j] = e; sum += e;
        }
      sum += __shfl_xor(sum, 16, 32);
      const float rs = 1.0f / sum;
      #pragma unroll
      for (int kt = 0; kt < 7; ++kt)
        #pragma unroll
        for (int j = 0; j < 8; ++j)
          s_a[(kt * 16 + j + half * 8) * AS + lcol] = lg[kt][j] * rs;
    }
    __syncthreads();   // (C)

    // ---- agg GEMM (a[112xTL] * x^T[TLx80]); wave wv<7 owns cluster tile kt=wv ----
    if (wv < 7) {
      const int k0 = wv * 16;
      for (int ls = 0; ls < TL / 4; ++ls) {         // inner dim l, 4 per WMMA
        const int lb = ls * 4 + half * 2;
        v2f af;
        af.x = s_a[(k0 + m16) * AS + lb];
        af.y = s_a[(k0 + m16) * AS + lb + 1];
        #pragma unroll
        for (int ct = 0; ct < 5; ++ct) {
          v2f bf;
          bf.x = xc[(ct * 16 + m16) * TLP + lb];     // x^T[lb][c]   = x[c][lb]
          bf.y = xc[(ct * 16 + m16) * TLP + lb + 1]; // x^T[lb+1][c] = x[c][lb+1]
          accA[ct] = __builtin_amdgcn_wmma_f32_16x16x4_f32(
              false, af, false, bf, (short)0, accA[ct], false, false);
        }
      }
    }
    __syncthreads();   // (D)
  }

  // ---- dump agg accumulators to LDS (stride AGS reuse of s_a) ----
  if (wv < 7) {
    #pragma unroll
    for (int ct = 0; ct < 5; ++ct)
      #pragma unroll
      for (int j = 0; j < 8; ++j)
        s_a[(wv * 16 + j + half * 8) * AGS + ct * 16 + m16] = accA[ct][j];
  }
  __syncthreads();

  // ---- residual vs centroids + intra-normalize per cluster ----
  if (tid < KREAL) {
    const int k = tid;
    const float asum = s_a[k * AGS + 64];            // ones-row = sum_l a[k,l]
    float ss = 0.0f;
    for (int c = 0; c < CC; ++c) {
      float v = s_a[k * AGS + c] - cents[k * CC + c] * asum;
      s_a[k * AGS + c] = v;
      ss += v * v;
    }
    const float ri = 1.0f / fmaxf(sqrtf(ss), 1e-12f);
    float part = 0.0f;
    for (int c = 0; c < CC; ++c) {
      float v = s_a[k * AGS + c] * ri;
      s_a[k * AGS + c] = v;
      part += v * v;
    }
    atomicAdd(&s_gss, part);
  }
  __syncthreads();

  // ---- global L2 normalize and write out ----
  const float gs = 1.0f / fmaxf(sqrtf(s_gss), 1e-12f);
  for (int idx = tid; idx < KREAL * CC; idx += 256)
    out[(size_t)n * (KREAL * CC) + idx] = s_a[(idx >> 6) * AGS + (idx & 63)] * gs;
}

extern "C" void kernel_launch(void* const* d_in, const int* in_sizes, int n_in,
                              void* d_out, int out_size, void* d_ws, size_t ws_size,
                              hipStream_t stream) {
  (void)n_in; (void)out_size; (void)d_ws; (void)ws_size;
  const float* x      = (const float*)d_in[0];   // [N,64,64,64]
  const float* conv_w = (const float*)d_in[1];   // [100,64]
  const float* conv_b = (const float*)d_in[2];   // [100]
  const float* cents  = (const float*)d_in[3];   // [100,64]
  float* out = (float*)d_out;                    // [N,6400]
  const int N = in_sizes[0] / (CC * LTOT);
  netvlad_fused<<<dim3(N), dim3(256), 0, stream>>>(x, conv_w, conv_b, cents, out);
}